// EMDLosspy_9990093931231
// MI455X (gfx1250) — compile-verified
//
#include <hip/hip_runtime.h>
#include <math.h>

// ---------------- problem constants (from reference) ----------------
#define BATCH   16
#define NPTS    2048
#define EPS_F   0.02f
#define INV_EPS 50.0f          // 1/EPS
#define N_ITERS 50
#define TPB     256
#define NWAVES  (TPB / 32)
#define NTILES  (NPTS / 16)    // 128 tiles of 16
#define LOG_MARG (-7.62461899f) // -log(2048)

typedef __attribute__((ext_vector_type(2))) float v2f;
typedef __attribute__((ext_vector_type(8))) float v8f;

// D = A(16x4) * B(4x16) + 0, f32 WMMA
__device__ __forceinline__ v8f wmma_dot(v2f a, v2f b) {
    v8f c = {};
    return __builtin_amdgcn_wmma_f32_16x16x4_f32(
        /*neg_a=*/false, a, /*neg_b=*/false, b,
        /*c_mod=*/(short)0, c, /*reuse_a=*/false, /*reuse_b=*/false);
}

// Build packed column planes for a Sinkhorn half-update:
//   sCol[4i+0..2] = 2*x, 2*y, 2*z ;  sCol[4i+3] = t_i = dual_i - |y_i|^2
// so that WMMA with A = [x, y, z, 1] yields D = 2*x.y + dual_j - |y_j|^2,
// i.e. the logit numerator (dual_j - C_ij) plus |x_i|^2 (subtracted later).
__device__ __forceinline__ void load_cols_scaled(const float* __restrict__ pts,
                                                 const float* dual, float* sCol) {
    for (int i = threadIdx.x; i < NPTS; i += TPB) {
        const float x = pts[3 * i + 0];
        const float y = pts[3 * i + 1];
        const float z = pts[3 * i + 2];
        const float yy = x * x + y * y + z * z;
        sCol[4 * i + 0] = 2.0f * x;
        sCol[4 * i + 1] = 2.0f * y;
        sCol[4 * i + 2] = 2.0f * z;
        sCol[4 * i + 3] = dual[i] - yy;
    }
}

// Raw planes for the final P*C pass: coords + g, plus |y|^2 side array.
__device__ __forceinline__ void load_cols_raw(const float* __restrict__ pts,
                                              const float* g, float* sCol,
                                              float* sYY) {
    for (int i = threadIdx.x; i < NPTS; i += TPB) {
        const float x = pts[3 * i + 0];
        const float y = pts[3 * i + 1];
        const float z = pts[3 * i + 2];
        sCol[4 * i + 0] = x;
        sCol[4 * i + 1] = y;
        sCol[4 * i + 2] = z;
        sCol[4 * i + 3] = g[i];
        sYY[i] = x * x + y * y + z * z;
    }
}

// One Sinkhorn half-update:
//   dualOut_i = EPS * (logMarg - logsumexp_j((dualCol_j - C_ij)/EPS))
// Rows from global (cache-resident), columns pre-fused in LDS (sCol).
__device__ __forceinline__ void half_update(
    const float* __restrict__ rowPts,   // global [NPTS*3]
    const float* sCol,                  // LDS packed {2x,2y,2z,t}
    float* dualOut,                     // LDS
    float logMarg) {

    const int lane = threadIdx.x & 31;
    const int wave = threadIdx.x >> 5;
    const int hi   = lane >> 4;
    const int lr   = lane & 15;

    for (int I = wave; I < NTILES; I += NWAVES) {
        const int rbase = I * 16;

        // A operand (16x4): lanes 0-15 hold K=0,1 = (x,y);
        // lanes 16-31 hold K=2,3 = (z, 1)  -> K=3 row carries the bias.
        v2f A;
        {
            const float* p = rowPts + (rbase + lr) * 3;
            A.x = hi ? p[2] : p[0];
            A.y = hi ? 1.0f : p[1];
        }
        float xx[8];
#pragma unroll
        for (int r = 0; r < 8; ++r) {
            const float* p = rowPts + (rbase + hi * 8 + r) * 3;
            xx[r] = p[0] * p[0] + p[1] * p[1] + p[2] * p[2];
        }

        // ---- pass 1: row maxima of D (logit is monotone in D) ----
        float mxD[8];
#pragma unroll
        for (int r = 0; r < 8; ++r) mxD[r] = -3.0e38f;

        for (int J = 0; J < NTILES; ++J) {
            const int col = J * 16 + lr;
            const float2 bv =
                *reinterpret_cast<const float2*>(&sCol[4 * col + hi * 2]);
            v2f Bv; Bv.x = bv.x; Bv.y = bv.y;
            const v8f d = wmma_dot(A, Bv);
#pragma unroll
            for (int r = 0; r < 8; ++r) mxD[r] = fmaxf(mxD[r], d[r]);
        }
#pragma unroll
        for (int r = 0; r < 8; ++r) {
#pragma unroll
            for (int k = 1; k <= 8; k <<= 1)
                mxD[r] = fmaxf(mxD[r], __shfl_xor(mxD[r], k, 32));
        }

        float msc[8], s[8];
#pragma unroll
        for (int r = 0; r < 8; ++r) { msc[r] = mxD[r] * INV_EPS; s[r] = 0.0f; }

        // ---- pass 2: sum exp((D - maxD)/eps) ----
        for (int J = 0; J < NTILES; ++J) {
            const int col = J * 16 + lr;
            const float2 bv =
                *reinterpret_cast<const float2*>(&sCol[4 * col + hi * 2]);
            v2f Bv; Bv.x = bv.x; Bv.y = bv.y;
            const v8f d = wmma_dot(A, Bv);
#pragma unroll
            for (int r = 0; r < 8; ++r)
                s[r] += __expf(d[r] * INV_EPS - msc[r]);
        }
#pragma unroll
        for (int r = 0; r < 8; ++r) {
#pragma unroll
            for (int k = 1; k <= 8; k <<= 1)
                s[r] += __shfl_xor(s[r], k, 32);
            if (lr == r) {
                const float lse = (mxD[r] - xx[r]) * INV_EPS + __logf(s[r]);
                dualOut[rbase + hi * 8 + r] = EPS_F * (logMarg - lse);
            }
        }
    }
}

__global__ void __launch_bounds__(TPB)
sinkhorn_emd_kernel(const float* __restrict__ tmpl,
                    const float* __restrict__ src,
                    float* __restrict__ perBatch) {
    __shared__ float sCol[NPTS * 4];   // packed column planes
    __shared__ float sF[NPTS], sG[NPTS], sYY[NPTS];
    __shared__ float sRed[NWAVES];

    const float* tb = tmpl + (size_t)blockIdx.x * NPTS * 3;
    const float* sb = src  + (size_t)blockIdx.x * NPTS * 3;

    for (int i = threadIdx.x; i < NPTS; i += TPB) { sF[i] = 0.0f; sG[i] = 0.0f; }
    __syncthreads();

    for (int it = 0; it < N_ITERS; ++it) {
        load_cols_scaled(sb, sG, sCol);
        __syncthreads();
        half_update(tb, sCol, sF, LOG_MARG);   // update f
        __syncthreads();
        load_cols_scaled(tb, sF, sCol);
        __syncthreads();
        half_update(sb, sCol, sG, LOG_MARG);   // update g
        __syncthreads();
    }

    // ---- final: sum_ij exp((f_i + g_j - C_ij)/eps) * C_ij ----
    load_cols_raw(sb, sG, sCol, sYY);
    __syncthreads();

    const int lane = threadIdx.x & 31;
    const int wave = threadIdx.x >> 5;
    const int hi   = lane >> 4;
    const int lr   = lane & 15;

    float acc = 0.0f;
    for (int I = wave; I < NTILES; I += NWAVES) {
        const int rbase = I * 16;
        v2f A;   // K=3 slot zero: D = x . y (plane 3 = g multiplied by 0)
        {
            const float* p = tb + (rbase + lr) * 3;
            A.x = hi ? p[2] : p[0];
            A.y = hi ? 0.0f : p[1];
        }
        float xx[8], fr[8];
#pragma unroll
        for (int r = 0; r < 8; ++r) {
            const float* p = tb + (rbase + hi * 8 + r) * 3;
            xx[r] = p[0] * p[0] + p[1] * p[1] + p[2] * p[2];
            fr[r] = sF[rbase + hi * 8 + r];
        }
        for (int J = 0; J < NTILES; ++J) {
            const int col = J * 16 + lr;
            const float2 bv =
                *reinterpret_cast<const float2*>(&sCol[4 * col + hi * 2]);
            v2f Bv; Bv.x = bv.x; Bv.y = bv.y;
            const float gc = sCol[4 * col + 3];
            const float yy = sYY[col];
            const v8f d = wmma_dot(A, Bv);
#pragma unroll
            for (int r = 0; r < 8; ++r) {
                const float cost = xx[r] + yy - 2.0f * d[r];
                acc += __expf((fr[r] + gc - cost) * INV_EPS) * cost;
            }
        }
    }
#pragma unroll
    for (int k = 1; k <= 16; k <<= 1) acc += __shfl_xor(acc, k, 32);
    if (lane == 0) sRed[wave] = acc;
    __syncthreads();
    if (threadIdx.x == 0) {
        float t = 0.0f;
#pragma unroll
        for (int w = 0; w < NWAVES; ++w) t += sRed[w];
        perBatch[blockIdx.x] = t;
    }
}

__global__ void finalize_kernel(const float* __restrict__ perBatch,
                                float* __restrict__ out) {
    if (threadIdx.x == 0) {
        float t = 0.0f;
#pragma unroll
        for (int b = 0; b < BATCH; ++b) t += perBatch[b];
        out[0] = t / (float)BATCH;  // mean over batch of sum(P*C)
    }
}

extern "C" void kernel_launch(void* const* d_in, const int* in_sizes, int n_in,
                              void* d_out, int out_size, void* d_ws, size_t ws_size,
                              hipStream_t stream) {
    (void)in_sizes; (void)n_in; (void)out_size; (void)ws_size;
    const float* tmpl = (const float*)d_in[0];
    const float* src  = (const float*)d_in[1];
    float* out = (float*)d_out;
    float* ws  = (float*)d_ws;   // 16 per-batch partial sums

    sinkhorn_emd_kernel<<<BATCH, TPB, 0, stream>>>(tmpl, src, ws);
    finalize_kernel<<<1, 32, 0, stream>>>(ws, out);
}